// HashedGrid_75436805587558
// MI455X (gfx1250) — compile-verified
//
#include <hip/hip_runtime.h>
#include <stdint.h>

// Hashed-grid embedding lookup for MI455X (gfx1250, wave32).
//
// h = (x0 ^ x1*19349663 ^ x2*83492791) & (2^22 - 1)      [exact: see analysis]
// out[n] = table[h[n]]  (rows of 4 float32 = 16B)
//
// Pure bandwidth kernel: idx/out are streamed with non-temporal hints so the
// 64 MB table stays resident in the 192 MB L2; all accesses are b128.

typedef __attribute__((ext_vector_type(4))) unsigned int uint32x4;
typedef __attribute__((ext_vector_type(4))) float        floatx4;

#define HASH_P1   19349663u
#define HASH_P2   83492791u
#define HASH_MASK 0x3FFFFFu   // NUM_ENTRIES = 2^22

__global__ __launch_bounds__(256) void hashed_grid_gather(
    const uint32x4* __restrict__ idx128,  // idx viewed as aligned 16B chunks
    const floatx4*  __restrict__ table,   // [2^22] rows of 4 floats
    floatx4*        __restrict__ out,     // [N] rows of 4 floats
    unsigned int npairs)                  // N/2 point-pairs
{
    unsigned int t = blockIdx.x * blockDim.x + threadIdx.x;
    if (t >= npairs) return;

    // Two points = 6 int64 coords = 48 bytes = three b128 loads (16B aligned:
    // byte offset 48*t). Streaming data -> non-temporal so L2 keeps the table.
    const uint32x4* p = idx128 + 3u * t;
    uint32x4 v0 = __builtin_nontemporal_load(p + 0); // {x0.lo,x0.hi, x1.lo,x1.hi}
    uint32x4 v1 = __builtin_nontemporal_load(p + 1); // {x2.lo,x2.hi, y0.lo,y0.hi}
    uint32x4 v2 = __builtin_nontemporal_load(p + 2); // {y1.lo,y1.hi, y2.lo,y2.hi}

    // 32-bit hash (exact reduction of the int64 reference math).
    unsigned int ha = (v0.x ^ (v0.z * HASH_P1) ^ (v1.x * HASH_P2)) & HASH_MASK;
    unsigned int hb = (v1.z ^ (v2.x * HASH_P1) ^ (v2.z * HASH_P2)) & HASH_MASK;

    // Random 16B gathers: default (RT) temporal policy -> table becomes and
    // stays L2-resident (64 MB < 192 MB L2).
    floatx4 ra = table[ha];
    floatx4 rb = table[hb];

    // Streaming output: non-temporal b128 stores, 32B contiguous per thread.
    __builtin_nontemporal_store(ra, out + 2u * t + 0);
    __builtin_nontemporal_store(rb, out + 2u * t + 1);
}

// Scalar tail for an odd point count (not hit for N=8,000,000; kept for
// generality — launch decision is a pure function of in_sizes, so it is
// graph-capture safe and deterministic).
__global__ void hashed_grid_tail(
    const long long* __restrict__ idx,
    const floatx4*   __restrict__ table,
    floatx4*         __restrict__ out,
    unsigned int p)
{
    size_t base = (size_t)3 * p;
    unsigned int x0 = (unsigned int)idx[base + 0];
    unsigned int x1 = (unsigned int)idx[base + 1];
    unsigned int x2 = (unsigned int)idx[base + 2];
    unsigned int h  = (x0 ^ (x1 * HASH_P1) ^ (x2 * HASH_P2)) & HASH_MASK;
    out[p] = table[h];
}

extern "C" void kernel_launch(void* const* d_in, const int* in_sizes, int n_in,
                              void* d_out, int out_size, void* d_ws, size_t ws_size,
                              hipStream_t stream) {
    (void)n_in; (void)out_size; (void)d_ws; (void)ws_size;

    const uint32x4* idx128 = (const uint32x4*)d_in[0];  // int64 idx, reinterpreted
    const floatx4*  table  = (const floatx4*)d_in[1];   // float32 [2^22, 4]
    floatx4*        out    = (floatx4*)d_out;           // float32 [N, 4]

    long long n_points = (long long)in_sizes[0] / 3;    // 8,000,000
    unsigned int npairs = (unsigned int)(n_points / 2); // 4,000,000

    const int threads = 256;                            // 8 wave32 waves/block
    int blocks = (int)((npairs + threads - 1) / threads);
    if (blocks > 0) {
        hashed_grid_gather<<<blocks, threads, 0, stream>>>(idx128, table, out, npairs);
    }
    if (n_points & 1) {
        hashed_grid_tail<<<1, 1, 0, stream>>>(
            (const long long*)d_in[0], table, out, (unsigned int)(n_points - 1));
    }
}